// ConvCaps_23562190586559
// MI455X (gfx1250) — compile-verified
//
#include <hip/hip_runtime.h>
#include <math.h>

typedef __attribute__((ext_vector_type(2))) float v2f;
typedef __attribute__((ext_vector_type(4))) float v4f;
typedef __attribute__((ext_vector_type(8))) float v8f;

#define BK    144      // B*K*K input slots
#define NC    16       // output capsules C
#define PS    16       // pose size 4x4
#define OHW   144      // 12*12 output pixels
#define NPIX  196      // 14*14 input pixels
#define RITER 3
#define EPSV  1e-8f
#define NSLOT 9        // BK*NC/256 logit rows per thread

// LDS map (float offsets). Total = 39,568 floats = 158,272 B
// -> 2 workgroups resident per 320 KB WGP (16 waves, 4 per SIMD32).
// XS (im2col patch) is dead after the vote phase; the routing c-buffer aliases it.
// Logits live entirely in VGPRs (9 per thread); softmax/squash use 16-lane shuffles.
#define U_OFF   0                       // votes u[B][C][16]      36864
#define XS_OFF  (BK*NC*PS)              // patch xs[B][16]         2304
#define CB_OFF  XS_OFF                  // c[B][C]  (aliases XS)   2304
#define AI_OFF  (XS_OFF + BK*PS)        // a_in[B]                  144
#define VB_OFF  (AI_OFF + BK)           // v[C][16]                 256
#define SMEM_FLOATS (VB_OFF + 256)

extern "C" __global__ void __launch_bounds__(256)
convcaps_fused(const float* __restrict__ x,
               const float* __restrict__ a,
               const float* __restrict__ Wg,
               float* __restrict__ pose,
               float* __restrict__ act)
{
  extern __shared__ float sm[];
  const int tid = threadIdx.x;
  const int bb  = blockIdx.x / OHW;
  const int hw  = blockIdx.x % OHW;
  const int h   = hw / 12, w = hw - (hw / 12) * 12;

  // W (147 KB) is reused by all 2304 blocks -> keep it hot in L2
  __builtin_prefetch(Wg + tid * 144, 0, 0);

  // ---- phase 0: stage im2col patch + activations ----
  for (int idx = tid; idx < BK * PS; idx += 256) {
    int Bk = idx >> 4, p = idx & 15;
    int Bin = Bk / 9, r = Bk - Bin * 9, kh = r / 3, kw = r - kh * 3;
    sm[XS_OFF + idx] =
        x[(((size_t)bb * 16 + Bin) * 16 + p) * NPIX + (h + kh) * 14 + (w + kw)];
  }
  for (int idx = tid; idx < BK; idx += 256) {
    int Bin = idx / 9, r = idx - Bin * 9, kh = r / 3, kw = r - kh * 3;
    sm[AI_OFF + idx] = a[((size_t)bb * 16 + Bin) * NPIX + (h + kh) * 14 + (w + kw)];
  }
  __syncthreads();

  // ---- phase 1: votes u[B][C][q] = xp[B] @ W[B][C], q = k*4 + i ----
  // WMMA f32 16x16x4: A rows 0-3 = xp (i x j), B 4x16 = W[B] for one
  // 4-capsule group (N = C'*4 + k), K = j.  W is double-buffered in VGPRs so
  // the 8-load clause for B+8 stays in flight under the 4 WMMAs (partial waits).
  {
    const int wv   = tid >> 5;
    const int lane = tid & 31;
    const int half = lane >> 4;      // 0: K=0,1  1: K=2,3 (ISA 16x4 f32 layout)
    const int row  = lane & 15;      // A: M row / B: N column
    const int jj   = half << 1;
    const float* wbase = Wg + ((row >> 2) << 4) + (jj << 2) + (row & 3);

    v2f wcur[4], wnxt[4];
#pragma unroll
    for (int cg = 0; cg < 4; ++cg) {
      const float* wp = wbase + wv * 256 + cg * 64;
      wcur[cg].x = wp[0]; wcur[cg].y = wp[4];        // j, j+1 (stride 4 floats)
    }
    for (int B = wv; B < BK; B += 8) {
      const int Bn = (B + 8 < BK) ? (B + 8) : wv;    // harmless redundant tail load
#pragma unroll
      for (int cg = 0; cg < 4; ++cg) {
        const float* wp = wbase + Bn * 256 + cg * 64;
        wnxt[cg].x = wp[0]; wnxt[cg].y = wp[4];
      }
      v2f A = {0.f, 0.f};
      if (row < 4) {                 // only i=0..3 rows are real
        const float* xr = &sm[XS_OFF + B * 16 + (row << 2) + jj];
        A.x = xr[0]; A.y = xr[1];
      }
#pragma unroll
      for (int cg = 0; cg < 4; ++cg) {
        const int C = (cg << 2) + (row >> 2);        // N = C'*4 + k
        v8f Cz = {};
        v8f D = __builtin_amdgcn_wmma_f32_16x16x4_f32(
                    false, A, false, wcur[cg], (short)0, Cz, false, false);
        if (half == 0) {             // lanes 0-15 hold M=0..3 in D[0..3]
          v4f st = { D[0], D[1], D[2], D[3] };       // q = k*4 + i, i=0..3
          *(v4f*)&sm[U_OFF + ((B * 16 + C) << 4) + ((row & 3) << 2)] = st;
        }
      }
#pragma unroll
      for (int cg = 0; cg < 4; ++cg) wcur[cg] = wnxt[cg];
    }
  }
  __syncthreads();

  // ---- phase 2: dynamic routing; logits in registers, 2 barriers/iter ----
  const int Cc = tid >> 4;           // this thread's (C, q) for s/v
  const int qq = tid & 15;
  float L[NSLOT];                    // logits[j*16+Cc][qq]
#pragma unroll
  for (int j = 0; j < NSLOT; ++j) L[j] = 0.f;
  float vv = 0.f;

  for (int it = 0; it < RITER; ++it) {
    // c[B][C] = softmax_C(logits[B][:])[C] * a_in[B]; row B = j*16+Cc, col C = qq
    if (it == 0) {                   // logits all zero -> uniform 1/16
#pragma unroll
      for (int j = 0; j < NSLOT; ++j)
        sm[CB_OFF + j * 256 + tid] = sm[AI_OFF + j * 16 + Cc] * (1.f / 16.f);
    } else {
#pragma unroll
      for (int j = 0; j < NSLOT; ++j) {
        float l = L[j];
        float mx = l;
#pragma unroll
        for (int m = 8; m >= 1; m >>= 1) mx = fmaxf(mx, __shfl_xor(mx, m, 16));
        float e = expf(l - mx);
        float ssum = e;
#pragma unroll
        for (int m = 8; m >= 1; m >>= 1) ssum += __shfl_xor(ssum, m, 16);
        sm[CB_OFF + j * 256 + tid] = e * sm[AI_OFF + j * 16 + Cc] / ssum;
      }
    }
    __syncthreads();

    // s[C][q] = sum_B c[B][C] * u[B][C][q]  (u reads lane-consecutive; 4 chains)
    float s0 = 0.f, s1 = 0.f, s2 = 0.f, s3 = 0.f;
    for (int B = 0; B < BK; B += 4) {
      s0 += sm[CB_OFF + ((B    ) << 4) + Cc] * sm[U_OFF + ((B    ) << 8) + tid];
      s1 += sm[CB_OFF + ((B + 1) << 4) + Cc] * sm[U_OFF + ((B + 1) << 8) + tid];
      s2 += sm[CB_OFF + ((B + 2) << 4) + Cc] * sm[U_OFF + ((B + 2) << 8) + tid];
      s3 += sm[CB_OFF + ((B + 3) << 4) + Cc] * sm[U_OFF + ((B + 3) << 8) + tid];
    }
    float s = (s0 + s1) + (s2 + s3);

    // squash: ||s||^2 over the 16-lane capsule group
    float n2 = s * s;
#pragma unroll
    for (int m = 8; m >= 1; m >>= 1) n2 += __shfl_xor(n2, m, 16);
    vv = s * n2 / ((1.f + n2) * sqrtf(n2 + EPSV));
    sm[VB_OFF + tid] = vv;
    __syncthreads();

    // logits[B][C] (+)= <u[B][C][:], v[C][:]>  (register update; dead on last it)
    if (it < RITER - 1) {
      v4f vr0 = *(const v4f*)&sm[VB_OFF + (qq << 4) + 0];
      v4f vr1 = *(const v4f*)&sm[VB_OFF + (qq << 4) + 4];
      v4f vr2 = *(const v4f*)&sm[VB_OFF + (qq << 4) + 8];
      v4f vr3 = *(const v4f*)&sm[VB_OFF + (qq << 4) + 12];
#pragma unroll
      for (int j = 0; j < NSLOT; ++j) {
        const v4f* ur = (const v4f*)&sm[U_OFF + ((tid + (j << 8)) << 4)];
        v4f u0 = ur[0], u1 = ur[1], u2 = ur[2], u3 = ur[3];
        float acc = u0.x * vr0.x + u0.y * vr0.y + u0.z * vr0.z + u0.w * vr0.w
                  + u1.x * vr1.x + u1.y * vr1.y + u1.z * vr1.z + u1.w * vr1.w
                  + u2.x * vr2.x + u2.y * vr2.y + u2.z * vr2.z + u2.w * vr2.w
                  + u3.x * vr3.x + u3.y * vr3.y + u3.z * vr3.z + u3.w * vr3.w;
        L[j] = (it == 0) ? acc : (L[j] + acc);
      }
      // no barrier needed: next CB writes are fenced by the barrier after them
    }
  }

  // ---- outputs: pose (b,C,p,h,w) with p = i*4+k, act (b,C,h,w) ----
  {
    const int qmap = ((qq & 3) << 2) | (qq >> 2);      // output p -> internal q
    pose[(((size_t)bb * 16 + Cc) * 16 + qq) * OHW + hw] =
        sm[VB_OFF + (Cc << 4) + qmap];
    float n2v = vv * vv;                               // ||v[Cc]||^2 via shuffles
#pragma unroll
    for (int m = 8; m >= 1; m >>= 1) n2v += __shfl_xor(n2v, m, 16);
    if (qq == 0)
      act[((size_t)bb * 16 + Cc) * OHW + hw] = sqrtf(n2v + EPSV);
  }
}

extern "C" void kernel_launch(void* const* d_in, const int* in_sizes, int n_in,
                              void* d_out, int out_size, void* d_ws, size_t ws_size,
                              hipStream_t stream) {
  const float* x  = (const float*)d_in[0];   // (16,16,16,14,14)
  const float* a  = (const float*)d_in[1];   // (16,16,14,14)
  const float* Wg = (const float*)d_in[2];   // (1,144,16,4,4)
  float* pose = (float*)d_out;               // 589824 floats
  float* actv = pose + 16 * 16 * 16 * OHW;   // + 36864 floats

  const size_t smem = (size_t)SMEM_FLOATS * sizeof(float);   // 158,272 B
  hipFuncSetAttribute((const void*)convcaps_fused,
                      hipFuncAttributeMaxDynamicSharedMemorySize, (int)smem);
  convcaps_fused<<<16 * OHW, 256, smem, stream>>>(x, a, Wg, pose, actv);
}